// LorentzSelfAttention_9560597201096
// MI455X (gfx1250) — compile-verified
//
#include <hip/hip_runtime.h>
#include <math.h>

// ---------------------------------------------------------------------------
// LorentzSelfAttention on gfx1250 (CDNA5, wave32, WMMA f32 16x16x4)
// B=2, T=512, D=512, H=8, N=32, A=33
// ---------------------------------------------------------------------------

typedef __attribute__((ext_vector_type(2))) float v2f;
typedef __attribute__((ext_vector_type(8))) float v8f;

#define WMMA4(a, b, c) \
  __builtin_amdgcn_wmma_f32_16x16x4_f32(false, (a), false, (b), (short)0, (c), false, false)

constexpr int kB = 2, kT = 512, kD = 512, kH = 8, kN = 32, kA = 33;
constexpr int kBH = kB * kH;          // 16
constexpr int kBT = kB * kT;          // 1024
constexpr int kHN = kH * kN;          // 256
constexpr int kHA = kH * kA;          // 264
constexpr int ASTR = 36;              // padded ambient stride (mult of 4)
constexpr int VSTR = 48;              // padded ambient stride for step GEMM B (mult of 16)
constexpr float EPSc   = 1e-6f;
constexpr float TINYc  = 1e-15f;
constexpr float MAXAc  = 1e6f;
constexpr float MAXDc  = 50.0f;

__device__ __forceinline__ float clampf(float x, float lo, float hi) {
  return fminf(fmaxf(x, lo), hi);
}

// ---------------------------------------------------------------------------
// Generic fp32 GEMM: C[M,N] = A[M,K] * B[K,N] (+bias), via v_wmma_f32_16x16x4_f32
// grid.y = M/16, grid.x covers N in chunks of 128 (8 waves x 16 cols), block=256
// A-frag (16x4): lane l -> row = l&15, ks = (l>>4)*2, elems K=ks, ks+1
// B-frag (4x16): lane l -> col = l&15, ks = (l>>4)*2, elems K=ks, ks+1
// D (16x16): vgpr v, lane l -> row = v + 8*(l>>4), col = l&15
// ---------------------------------------------------------------------------
__global__ void gemm16(const float* __restrict__ A, const float* __restrict__ B,
                       const float* __restrict__ bias, float* __restrict__ C,
                       int N, int K) {
  const int lane = threadIdx.x & 31;
  const int wv   = threadIdx.x >> 5;
  const int m    = lane & 15;
  const int kq   = (lane >> 4) << 1;
  const int colb = blockIdx.x * 128 + wv * 16;
  if (colb >= N) return;                       // wave-uniform
  const int col  = colb + m;
  const int rowt = blockIdx.y * 16;

  const float* Ar = A + (size_t)(rowt + m) * K + kq;   // contiguous pair per step
  const float* Bp = B + (size_t)kq * N + col;          // bump by 4*N per step
  const size_t bstep = (size_t)4 * N;

  v8f acc = {0.f, 0.f, 0.f, 0.f, 0.f, 0.f, 0.f, 0.f};
  #pragma unroll 4
  for (int k = 0; k < K; k += 4) {
    __builtin_prefetch(Bp + 8 * bstep, 0, 3);          // near-scope prefetch of B stream
    v2f a = *(const v2f*)(Ar + k);
    v2f b;
    b.x = Bp[0];
    b.y = Bp[N];
    Bp += bstep;
    acc = WMMA4(a, b, acc);
  }
  const float bv = bias ? bias[col] : 0.0f;
  const int rbase = rowt + ((lane >> 4) << 3);
  #pragma unroll
  for (int v = 0; v < 8; v++)
    C[(size_t)(rbase + v) * N + col] = acc[v] + bv;
}

// ---------------------------------------------------------------------------
// Lift: u = U + b (32-d tangent), expmap at origin, project to hyperboloid.
// Writes Q (stride 36), Km/Vm (time-negated, stride 36), Vp (stride 48),
// Lorentz self-norms nQ, nV. One thread per (b,h,t) element.
// ---------------------------------------------------------------------------
__device__ __forceinline__ void lift_one(const float* __restrict__ urow,
                                         const float* __restrict__ brow,
                                         float* __restrict__ y /* 33 */) {
  float u[kN];
  float ss = 0.f;
  #pragma unroll
  for (int i = 0; i < kN; i++) {
    u[i] = urow[i] + brow[i];
    ss += u[i] * u[i];
  }
  const float vn = sqrtf(fmaxf(ss, TINYc));
  const float ch = coshf(vn);
  const float sh = sinhf(vn) / fmaxf(vn, TINYc);
  float y0 = ch;
  float xx = -y0 * y0;
  #pragma unroll
  for (int i = 0; i < kN; i++) {
    y[i + 1] = sh * u[i];
    xx += y[i + 1] * y[i + 1];
  }
  const float sc = sqrtf(fmaxf(fabsf(xx), TINYc));
  y[0] = fabsf(y0 / sc);
  #pragma unroll
  for (int i = 0; i < kN; i++) y[i + 1] /= sc;
}

__global__ void lift_kernel(const float* __restrict__ Uq, const float* __restrict__ Uk,
                            const float* __restrict__ Uv,
                            const float* __restrict__ bq, const float* __restrict__ bk,
                            const float* __restrict__ bv,
                            float* __restrict__ Q, float* __restrict__ Km,
                            float* __restrict__ Vm, float* __restrict__ Vp,
                            float* __restrict__ nQ, float* __restrict__ nV) {
  const int g = blockIdx.x * blockDim.x + threadIdx.x;
  if (g >= kBH * kT) return;
  const int bh = g / kT, t = g % kT;
  const int b = bh / kH, h = bh % kH;
  const size_t urow = ((size_t)b * kT + t) * kHN + h * kN;
  float y[kA];

  // Q
  lift_one(Uq + urow, bq + h * kN, y);
  {
    float* o = Q + (size_t)g * ASTR;
    float nn = -y[0] * y[0];
    #pragma unroll
    for (int a = 0; a < kA; a++) o[a] = y[a];
    for (int a = kA; a < ASTR; a++) o[a] = 0.f;
    for (int a = 1; a < kA; a++) nn += y[a] * y[a];
    nQ[g] = nn;
  }
  // K -> Km (time component negated so S = dot(Q, Km) == mink(Q,K))
  lift_one(Uk + urow, bk + h * kN, y);
  {
    float* o = Km + (size_t)g * ASTR;
    o[0] = -y[0];
    #pragma unroll
    for (int a = 1; a < kA; a++) o[a] = y[a];
    for (int a = kA; a < ASTR; a++) o[a] = 0.f;
  }
  // V -> Vm (negated, stride 36) and Vp (plain, stride 48)
  lift_one(Uv + urow, bv + h * kN, y);
  {
    float* om = Vm + (size_t)g * ASTR;
    float* op = Vp + (size_t)g * VSTR;
    om[0] = -y[0];
    op[0] = y[0];
    float nn = -y[0] * y[0];
    #pragma unroll
    for (int a = 1; a < kA; a++) { om[a] = y[a]; op[a] = y[a]; nn += y[a] * y[a]; }
    for (int a = kA; a < ASTR; a++) om[a] = 0.f;
    for (int a = kA; a < VSTR; a++) op[a] = 0.f;
    nV[g] = nn;
  }
}

// ---------------------------------------------------------------------------
// Attention: one workgroup (4 waves / 128 thr) per (b,h, 16-row q-tile).
// Phase 1: WMMA  S1 = mink(Q,K) (16x512),  S2 = mink(Q,V) (16x512)  -> LDS
// Phase 2: softmax(-acosh^2) over S1; fold attn * acosh(a2)/||u|| into W (LDS),
//          accumulate c_q = sum W*a2 per row.
// Phase 3: WMMA  Step(16x48) = W(16x512) * Vp(512x48)
// ---------------------------------------------------------------------------
constexpr int LS = 520;  // padded LDS row stride (floats)

__global__ void attn_kernel(const float* __restrict__ Q, const float* __restrict__ Km,
                            const float* __restrict__ Vm, const float* __restrict__ Vp,
                            const float* __restrict__ nQ, const float* __restrict__ nV,
                            float* __restrict__ Step, float* __restrict__ cbuf) {
  extern __shared__ float smem[];
  float* lds1   = smem;                 // 16 x LS : S1 -> W
  float* lds2   = lds1 + 16 * LS;       // 16 x LS : S2
  float* red    = lds2 + 16 * LS;       // 16 x 8
  float* rowmax = red + 128;            // 16
  float* rowsum = rowmax + 16;          // 16

  const int bh    = blockIdx.y;
  const int qbase = blockIdx.x * 16;
  const int tid   = threadIdx.x;
  const int lane  = tid & 31;
  const int wv    = tid >> 5;
  const int m     = lane & 15;
  const int kq    = (lane >> 4) << 1;

  // ---- Phase 1: pairwise Lorentz inner products via WMMA (K-dim = 36) ----
  const float* Qr = Q + ((size_t)bh * kT + qbase + m) * ASTR + kq;
  v2f aq[9];
  #pragma unroll
  for (int ka = 0; ka < 9; ka++) aq[ka] = *(const v2f*)(Qr + 4 * ka);

  for (int tt = wv; tt < 64; tt += 4) {
    const int matv = tt >> 5;          // 0 -> S1 (Km), 1 -> S2 (Vm)
    const int ct   = tt & 31;          // column tile over k (0..31)
    const float* Br = (matv ? Vm : Km) + ((size_t)bh * kT + ct * 16 + m) * ASTR + kq;
    v8f acc = {0.f, 0.f, 0.f, 0.f, 0.f, 0.f, 0.f, 0.f};
    #pragma unroll
    for (int ka = 0; ka < 9; ka++) {
      v2f bfr = *(const v2f*)(Br + 4 * ka);
      acc = WMMA4(aq[ka], bfr, acc);
    }
    float* dst = matv ? lds2 : lds1;
    const int rb = (lane >> 4) << 3;
    #pragma unroll
    for (int v = 0; v < 8; v++) dst[(rb + v) * LS + ct * 16 + m] = acc[v];
  }
  __syncthreads();

  // ---- Phase 2: softmax + log-map weights (8 threads per q row) ----
  {
    const int row = tid >> 3, sub = tid & 7;
    const int c0 = sub * 64;
    const float* s1r = lds1 + row * LS;
    const float* s2r = lds2 + row * LS;
    const float nq = nQ[(size_t)bh * kT + qbase + row];
    const float* nvb = nV + (size_t)bh * kT;

    auto scoref = [](float S1v) {
      float a1 = clampf(-S1v, 1.0f + EPSc, MAXAc);
      float d  = fminf(acoshf(a1), MAXDc);
      return -d * d;                   // TAU = 1
    };

    float mx = -1e30f;
    for (int j = 0; j < 64; j++) mx = fmaxf(mx, scoref(s1r[c0 + j]));
    red[row * 8 + sub] = mx;
    __syncthreads();
    if (sub == 0) {
      float v = red[row * 8];
      for (int k2 = 1; k2 < 8; k2++) v = fmaxf(v, red[row * 8 + k2]);
      rowmax[row] = v;
    }
    __syncthreads();
    const float rm = rowmax[row];

    float se = 0.f;
    for (int j = 0; j < 64; j++) se += expf(scoref(s1r[c0 + j]) - rm);
    red[row * 8 + sub] = se;
    __syncthreads();
    if (sub == 0) {
      float v = 0.f;
      for (int k2 = 0; k2 < 8; k2++) v += red[row * 8 + k2];
      rowsum[row] = v;
    }
    __syncthreads();
    const float inv = 1.0f / rowsum[row];

    float cacc = 0.f;
    for (int j = 0; j < 64; j++) {
      const int col = c0 + j;
      const float attn = expf(scoref(s1r[col]) - rm) * inv;
      const float S2v = s2r[col];
      const float a2  = clampf(-S2v, 1.0f + EPSc, MAXAc);
      const float d2  = acoshf(a2);
      const float uu  = nvb[col] + 2.0f * a2 * S2v + a2 * a2 * nq;
      const float un  = sqrtf(fmaxf(uu, TINYc));
      const float w   = attn * d2 / fmaxf(un, TINYc);
      lds1[row * LS + col] = w;        // this thread owns [row][c0..c0+63]
      cacc += w * a2;
    }
    red[row * 8 + sub] = cacc;
    __syncthreads();
    if (sub == 0) {
      float v = 0.f;
      for (int k2 = 0; k2 < 8; k2++) v += red[row * 8 + k2];
      cbuf[(size_t)bh * kT + qbase + row] = v;
    }
  }
  __syncthreads();

  // ---- Phase 3: Step = W(16x512) x Vp(512x48) via WMMA ----
  if (wv < 3) {                        // wave-uniform; 3 N-tiles of 16 ambient cols
    const int abase = wv * 16;
    const float* Ap = lds1 + m * LS + kq;                       // LDS, 8B aligned
    const float* Bp = Vp + ((size_t)bh * kT + kq) * VSTR + abase + m;
    const size_t bstep = (size_t)4 * VSTR;
    v8f acc = {0.f, 0.f, 0.f, 0.f, 0.f, 0.f, 0.f, 0.f};
    #pragma unroll 4
    for (int kk = 0; kk < kT; kk += 4) {
      __builtin_prefetch(Bp + 8 * bstep, 0, 3);
      v2f a = *(const v2f*)(Ap + kk);
      v2f b;
      b.x = Bp[0];
      b.y = Bp[VSTR];
      Bp += bstep;
      acc = WMMA4(a, b, acc);
    }
    const int rb = (lane >> 4) << 3;
    #pragma unroll
    for (int v = 0; v < 8; v++)
      Step[((size_t)bh * kT + qbase + rb + v) * VSTR + abase + m] = acc[v];
  }
}

// ---------------------------------------------------------------------------
// Finalize: step += c*Q, norm clamp, expmap(Q, step), project, logmap(origin)
// -> Z_tan (B,T, H*A). One thread per (b,h,t).
// ---------------------------------------------------------------------------
__global__ void finalize_kernel(const float* __restrict__ Q, const float* __restrict__ Step,
                                const float* __restrict__ cbuf, float* __restrict__ Zt) {
  const int g = blockIdx.x * blockDim.x + threadIdx.x;
  if (g >= kBH * kT) return;
  const int bh = g / kT, t = g % kT;
  const int b = bh / kH, h = bh % kH;
  const float* q  = Q + (size_t)g * ASTR;
  const float* st = Step + (size_t)g * VSTR;
  const float cq  = cbuf[g];

  float s[kA];
  float sn = 0.f;
  #pragma unroll
  for (int a = 0; a < kA; a++) {
    s[a] = st[a] + cq * q[a];
    sn += (a == 0) ? -s[a] * s[a] : s[a] * s[a];
  }
  const float norm  = sqrtf(fmaxf(sn, TINYc));
  const float scale = fminf(norm, 20.0f) / (norm + 1e-9f);
  float sn2 = 0.f;
  #pragma unroll
  for (int a = 0; a < kA; a++) {
    s[a] *= scale;
    sn2 += (a == 0) ? -s[a] * s[a] : s[a] * s[a];
  }
  const float vn = sqrtf(fmaxf(sn2, TINYc));
  const float ch = coshf(vn);
  const float sh = sinhf(vn) / fmaxf(vn, TINYc);

  float y[kA];
  float xx = 0.f;
  #pragma unroll
  for (int a = 0; a < kA; a++) {
    y[a] = ch * q[a] + sh * s[a];
    xx += (a == 0) ? -y[a] * y[a] : y[a] * y[a];
  }
  const float sc = sqrtf(fmaxf(fabsf(xx), TINYc));
  #pragma unroll
  for (int a = 0; a < kA; a++) y[a] /= sc;
  y[0] = fabsf(y[0]);

  // log map at origin o = (1,0,...,0)
  const float al   = clampf(y[0], 1.0f + EPSc, MAXAc);
  const float dist = acoshf(al);
  const float u0   = y[0] + al;
  float uu = -u0 * u0;
  #pragma unroll
  for (int a = 1; a < kA; a++) uu += y[a] * y[a];
  const float un = sqrtf(fmaxf(uu, TINYc));
  const float f  = dist / fmaxf(un, TINYc);

  float* zr = Zt + ((size_t)b * kT + t) * kHA + h * kA;
  zr[0] = f * u0;
  #pragma unroll
  for (int a = 1; a < kA; a++) zr[a] = f * y[a];
}

// ---------------------------------------------------------------------------
// Host-side launcher
// ---------------------------------------------------------------------------
extern "C" void kernel_launch(void* const* d_in, const int* in_sizes, int n_in,
                              void* d_out, int out_size, void* d_ws, size_t ws_size,
                              hipStream_t stream) {
  (void)in_sizes; (void)n_in; (void)out_size; (void)ws_size;
  const float* x  = (const float*)d_in[0];
  const float* Wq = (const float*)d_in[1];
  const float* bq = (const float*)d_in[2];
  const float* Wk = (const float*)d_in[3];
  const float* bk = (const float*)d_in[4];
  const float* Wv = (const float*)d_in[5];
  const float* bv = (const float*)d_in[6];
  const float* Wo = (const float*)d_in[7];
  const float* bo = (const float*)d_in[8];
  float* out = (float*)d_out;
  float* ws  = (float*)d_ws;

  // workspace layout (floats); total ~2.75M floats (~10.5 MB)
  float* Uq = ws;                       // 1024*256
  float* Uk = Uq + (size_t)kBT * kHN;
  float* Uv = Uk + (size_t)kBT * kHN;
  float* Qh = Uv + (size_t)kBT * kHN;   // 8192*36
  float* Km = Qh + (size_t)kBH * kT * ASTR;
  float* Vm = Km + (size_t)kBH * kT * ASTR;
  float* Vp = Vm + (size_t)kBH * kT * ASTR;  // 8192*48
  float* nQ = Vp + (size_t)kBH * kT * VSTR;
  float* nV = nQ + (size_t)kBH * kT;
  float* cb = nV + (size_t)kBH * kT;
  float* St = cb + (size_t)kBH * kT;    // 8192*48
  float* Zt = St + (size_t)kBH * kT * VSTR;  // 1024*264

  // 1) QKV projections: U = X * W  (M=1024, K=512, N=256)
  dim3 g1((kHN + 127) / 128, kBT / 16);
  gemm16<<<g1, 256, 0, stream>>>(x, Wq, nullptr, Uq, kHN, kD);
  gemm16<<<g1, 256, 0, stream>>>(x, Wk, nullptr, Uk, kHN, kD);
  gemm16<<<g1, 256, 0, stream>>>(x, Wv, nullptr, Uv, kHN, kD);

  // 2) Lift to hyperboloid
  lift_kernel<<<(kBH * kT) / 256, 256, 0, stream>>>(Uq, Uk, Uv, bq, bk, bv,
                                                    Qh, Km, Vm, Vp, nQ, nV);

  // 3) Attention (scores + softmax + Karcher-step GEMM), 66 KB dynamic LDS
  const size_t shbytes = (size_t)(2 * 16 * LS + 128 + 32) * sizeof(float);
  attn_kernel<<<dim3(kT / 16, kBH), 128, shbytes, stream>>>(Qh, Km, Vm, Vp, nQ, nV, St, cb);

  // 4) exp-map + origin log-map -> Z_tan (B,T,264)
  finalize_kernel<<<(kBH * kT) / 256, 256, 0, stream>>>(Qh, St, cb, Zt);

  // 5) Output projection: Z = Z_tan * Wo + bo  (M=1024, K=264, N=512)
  gemm16<<<dim3(kD / 128, kBT / 16), 256, 0, stream>>>(Zt, Wo, bo, out, kD, kHA);
}